// customLoss_88510686036863
// MI455X (gfx1250) — compile-verified
//
#include <hip/hip_runtime.h>

typedef __attribute__((ext_vector_type(4))) float v4f;
typedef __attribute__((ext_vector_type(2))) float v2f;
typedef __attribute__((ext_vector_type(8))) float v8f;

// ---------------------------------------------------------------------------
// Stage 1: stream 805MB once; 8 lanes per row (4 rows per wave32).
// Non-temporal b128 loads; 3-step butterfly per row; deterministic block sum.
// ---------------------------------------------------------------------------
__global__ __launch_bounds__(256) void loss_stage1(const float* __restrict__ x,
                                                   float* __restrict__ partials,
                                                   int nRows, int totalWaves) {
  const int tid  = blockIdx.x * blockDim.x + threadIdx.x;
  const int wave = tid >> 5;
  const int lane = threadIdx.x & 31;
  const int grp  = lane >> 3;   // which of the 4 rows this wave-iteration
  const int g    = lane & 7;    // lane within the 8-lane row group

  float acc = 0.0f;
  for (int r = wave * 4 + grp; r < nRows; r += totalWaves * 4) {
    const float* p = x + (size_t)r * 384;
    float e2 = 0.0f, s12 = 0.0f, s23 = 0.0f, s13 = 0.0f;
#pragma unroll
    for (int j = 0; j < 4; ++j) {
      const int idx = g * 4 + j * 32;  // 8 lanes cover 128B contiguous per j
      v4f a = __builtin_nontemporal_load((const v4f*)(p + idx));        // v1
      v4f b = __builtin_nontemporal_load((const v4f*)(p + 128 + idx));  // v2
      v4f c = __builtin_nontemporal_load((const v4f*)(p + 256 + idx));  // v3
#pragma unroll
      for (int k = 0; k < 4; ++k) {
        const float va = a[k], vb = b[k], vc = c[k];
        const float m = (va + vc) * 0.5f;
        const float t = vb - m;  e2  += t * t;
        const float u = va - vb; s12 += u * u;
        const float w = vb - vc; s23 += w * w;
        const float y = va - vc; s13 += y * y;
      }
    }
    // reduce the 4 sums across this row's 8 lanes (3 butterfly steps)
#pragma unroll
    for (int off = 4; off >= 1; off >>= 1) {
      e2  += __shfl_xor(e2,  off, 32);
      s12 += __shfl_xor(s12, off, 32);
      s23 += __shfl_xor(s23, off, 32);
      s13 += __shfl_xor(s13, off, 32);
    }
    const float d = __builtin_sqrtf(s12) + __builtin_sqrtf(s23) + __builtin_sqrtf(s13);
    acc += e2 + 0.05f * d * d;   // all 8 lanes of the group add the same value
  }

  // full-wave butterfly: every lane's acc was duplicated 8x within its group,
  // so the 32-lane sum is exactly 8x the wave total -> exact *0.125 fixup.
#pragma unroll
  for (int off = 16; off >= 1; off >>= 1) acc += __shfl_xor(acc, off, 32);
  acc *= 0.125f;

  __shared__ float wsum[8];
  const int waveInBlock = threadIdx.x >> 5;
  if (lane == 0) wsum[waveInBlock] = acc;
  __syncthreads();
  if (threadIdx.x == 0) {
    float s = 0.0f;
#pragma unroll
    for (int i = 0; i < 8; ++i) s += wsum[i];  // fixed order -> deterministic
    partials[blockIdx.x] = s;
  }
}

// ---------------------------------------------------------------------------
// Stage 2: reduce block partials with V_WMMA_F32_16X16X4_F32 (B = ones).
// D[m][n] = sum_k A[m][k] + C -> column 0 of D holds the 16 row sums.
// Single full wave32; loads are unconditional (clamped address + cndmask
// masking) so EXEC stays all-ones with no branches around the WMMA.
// n must be >= 2 and even (launcher forces a multiple of 64).
// ---------------------------------------------------------------------------
__global__ __launch_bounds__(32) void loss_stage2(const float* __restrict__ partials,
                                                  float* __restrict__ result, int n) {
  const int lane = threadIdx.x;  // 0..31
  v8f c = {};
  v2f ones;
  ones[0] = 1.0f;
  ones[1] = 1.0f;

  for (int base = 0; base < n; base += 64) {
    const int i0 = base + lane * 2;
    const int cl = (i0 > n - 2) ? (n - 2) : i0;   // clamp: load stays in-bounds
    v2f a = *(const v2f*)(partials + cl);         // one global_load_b64 / lane
    a[0] = (i0     < n) ? a[0] : 0.0f;            // v_cndmask, no EXEC change
    a[1] = (i0 + 1 < n) ? a[1] : 0.0f;
    // D = A(16x4 f32) x ones(4x16 f32) + C   -> accumulates 64 values / issue
    c = __builtin_amdgcn_wmma_f32_16x16x4_f32(false, a, false, ones,
                                              (short)0, c, false, false);
  }

  // column N=0 of D: lane 0 holds M=0..7 in VGPRs 0..7, lane 16 holds M=8..15
  const float t  = c[0] + c[1] + c[2] + c[3] + c[4] + c[5] + c[6] + c[7];
  const float r0  = __shfl(t, 0, 32);
  const float r16 = __shfl(t, 16, 32);
  if (lane == 0) result[0] = r0 + r16;
}

// Fallback for a pathologically small workspace (not expected to run).
__global__ void loss_stage2_small(const float* __restrict__ partials,
                                  float* __restrict__ result, int n) {
  if (threadIdx.x == 0 && blockIdx.x == 0) {
    float s = 0.0f;
    for (int i = 0; i < n; ++i) s += partials[i];
    result[0] = s;
  }
}

// ---------------------------------------------------------------------------
extern "C" void kernel_launch(void* const* d_in, const int* in_sizes, int n_in,
                              void* d_out, int out_size, void* d_ws, size_t ws_size,
                              hipStream_t stream) {
  const float* x   = (const float*)d_in[0];
  float* res       = (float*)d_out;
  float* partials  = (float*)d_ws;

  const long total = (long)in_sizes[0];          // 524288 * 3 * 128
  const int nRows  = (int)(total / 384);

  const int cap = (int)(ws_size / sizeof(float));
  int nb = 1024;                                  // 1024 partials = 4KB of d_ws
  if (nb > cap) nb = cap & ~63;                   // keep a multiple of 64
  const int nt = 256;                             // 8 waves / block

  if (nb >= 64) {
    const int totalWaves = nb * nt / 32;
    loss_stage1<<<nb, nt, 0, stream>>>(x, partials, nRows, totalWaves);
    loss_stage2<<<1, 32, 0, stream>>>(partials, res, nb);
  } else {
    const int nbs = cap < 1 ? 1 : cap;            // degenerate-scratch fallback
    loss_stage1<<<nbs, nt, 0, stream>>>(x, partials, nRows, nbs * nt / 32);
    loss_stage2_small<<<1, 32, 0, stream>>>(partials, res, nbs);
  }
}